// RGCN_43533788512793
// MI455X (gfx1250) — compile-verified
//
#include <hip/hip_runtime.h>

#define NN 100000
#define EE 800000
#define DD 16
#define HH 128
#define OO 3
#define RR 2
#define LSLOPE 0.01f

typedef __attribute__((ext_vector_type(2))) float v2f;
typedef __attribute__((ext_vector_type(8))) float v8f;

// D = A(16x4) * B(4x16) + C, fp32, wave32. A: lane holds row (lane&15),
// ks = (lane>>4)*2+{0,1}. B: lane holds col (lane&15), same k split.
// C/D: vgpr i -> row i + 8*(lane>>4), col lane&15.
__device__ __forceinline__ v8f wmma4(v2f a, v2f b, v8f c) {
    return __builtin_amdgcn_wmma_f32_16x16x4_f32(
        /*neg_a=*/false, a, /*neg_b=*/false, b,
        /*c_mod=*/(short)0, c, /*reuse_a=*/false, /*reuse_b=*/false);
}

// ---------------- weight packing into WMMA B-fragment order ----------------
// Packed layout: P[kk*H + col] = {W[4*(kk>>1) + 2*(kk&1)][col],
//                                 W[4*(kk>>1) + 2*(kk&1) + 1][col]}
// so a lane's B fragment for k-step k0 is one float2 at (k0*2 + (lane>>4))*H + col.
__global__ void k_pack(const float* __restrict__ W, float2* __restrict__ P, int K) {
    const int idx = blockIdx.x * blockDim.x + threadIdx.x;  // over (K/2)*H entries
    if (idx >= (K / 2) * HH) return;
    const int col = idx & (HH - 1);
    const int kk  = idx >> 7;                 // 0 .. K/2-1
    const int k   = 4 * (kk >> 1) + 2 * (kk & 1);
    float2 p;
    p.x = W[k * HH + col];
    p.y = W[(k + 1) * HH + col];
    P[idx] = p;
}

// ---------------- edge statistics ----------------
__global__ void k_count(const int* __restrict__ dst, const int* __restrict__ et,
                        float* __restrict__ cnt) {
    int e = blockIdx.x * blockDim.x + threadIdx.x;
    if (e < EE) atomicAdd(&cnt[(size_t)et[e] * NN + dst[e]], 1.0f);
}

__global__ void k_inv(float* __restrict__ cnt) {
    int i = blockIdx.x * blockDim.x + threadIdx.x;
    if (i < RR * NN) cnt[i] = 1.0f / fmaxf(cnt[i], 1.0f);
}

// ---------------- input linear + leaky relu:  h = lrelu(x @ W1 + b1) ----------------
__global__ void k_input(const float* __restrict__ mf, const float* __restrict__ f,
                        const float2* __restrict__ P1, const float* __restrict__ b1,
                        float* __restrict__ h, int branch) {
    const int wave = (blockIdx.x * blockDim.x + threadIdx.x) >> 5;
    if (wave >= NN / 16) return;
    const int lane  = threadIdx.x & 31;
    const int row0  = wave * 16;
    const int arow  = row0 + (lane & 15);
    const int khalf = (lane >> 4) << 1;     // 0 or 2
    const int bhalf = lane >> 4;

    // Preload all 4 K-steps of A (K = D = 16)
    v2f afrag[4];
#pragma unroll
    for (int ks = 0; ks < 4; ++ks) {
        const int k = ks * 4 + khalf;
        float x0, x1;
        if (branch == 0) {
            x0 = mf[arow * DD + k];
            x1 = mf[arow * DD + k + 1];
        } else {
            x0 = f[arow * DD + k] - mf[arow * DD + k];
            x1 = f[arow * DD + k + 1] - mf[arow * DD + k + 1];
        }
        v2f a; a.x = x0; a.y = x1;
        afrag[ks] = a;
    }

    const int crow = row0 + ((lane >> 4) << 3);
#pragma unroll
    for (int ct = 0; ct < HH / 16; ++ct) {
        const int col = ct * 16 + (lane & 15);
        v8f c = {};
#pragma unroll
        for (int ks = 0; ks < 4; ++ks) {
            const float2 bv = P1[(ks * 2 + bhalf) * HH + col];
            v2f b; b.x = bv.x; b.y = bv.y;
            c = wmma4(afrag[ks], b, c);
        }
        const float bias = b1[col];
#pragma unroll
        for (int i = 0; i < 8; ++i) {
            float v = c[i] + bias;
            v = v > 0.f ? v : LSLOPE * v;
            h[(size_t)(crow + i) * HH + col] = v;
        }
    }
}

// ---------------- scatter-add of source features into per-relation aggregates ----------------
__global__ void k_scatter(const float* __restrict__ x, const int* __restrict__ src,
                          const int* __restrict__ dst, const int* __restrict__ et,
                          float* __restrict__ agg) {
    const int e = (blockIdx.x * blockDim.x + threadIdx.x) >> 5; // one wave per edge
    if (e >= EE) return;
    const int lane = threadIdx.x & 31;
    const int s = src[e], d = dst[e], t = et[e];
    const float4 v = *(const float4*)(&x[(size_t)s * HH + lane * 4]);
    float* o = agg + (size_t)t * NN * HH + (size_t)d * HH + lane * 4;
    atomicAdd(o + 0, v.x);
    atomicAdd(o + 1, v.y);
    atomicAdd(o + 2, v.z);
    atomicAdd(o + 3, v.w);
}

// ---------------- fused RGCN conv GEMM:
//   out = x @ wroot + (inv0*agg0) @ wrel0 + (inv1*agg1) @ wrel1 + bias
// Weights come pre-packed in B-fragment order (float2 per lane per k-step).
__global__ void k_conv(const float* __restrict__ x,      // [N,H]
                       const float* __restrict__ agg,    // [2,N,H]
                       const float2* __restrict__ Proot, // packed [H/2*H]
                       const float2* __restrict__ Prel0, // packed
                       const float2* __restrict__ Prel1, // packed
                       const float* __restrict__ bias,   // [H]
                       const float* __restrict__ inv,    // [2,N] reciprocals
                       float* __restrict__ out) {        // [N,H]
    const int wave = (blockIdx.x * blockDim.x + threadIdx.x) >> 5;
    if (wave >= NN / 16) return;
    const int lane  = threadIdx.x & 31;
    const int row0  = wave * 16;
    const int arow  = row0 + (lane & 15);
    const int khalf = (lane >> 4) << 1;
    const int bhalf = lane >> 4;

    const float s1 = inv[arow];          // relation 0 mean scale for this A-row
    const float s2 = inv[NN + arow];     // relation 1

    v8f c[HH / 16];
#pragma unroll
    for (int t = 0; t < HH / 16; ++t) c[t] = (v8f){};

    const float*  As[3] = { x, agg, agg + (size_t)NN * HH };
    const float2* Ps[3] = { Proot, Prel0, Prel1 };
    const float   sc[3] = { 1.0f, s1, s2 };

    for (int si = 0; si < 3; ++si) {
        const float*  A = As[si] + (size_t)arow * HH;
        const float2* P = Ps[si];
        const float   s = sc[si];
#pragma unroll 4
        for (int k0 = 0; k0 < HH / 4; ++k0) {
            const float2 av = *(const float2*)(&A[k0 * 4 + khalf]);
            v2f a; a.x = av.x * s; a.y = av.y * s;
            const float2* Pk = P + (k0 * 2 + bhalf) * HH;
#pragma unroll
            for (int t = 0; t < HH / 16; ++t) {
                const float2 bv = Pk[t * 16 + (lane & 15)];
                v2f b; b.x = bv.x; b.y = bv.y;
                c[t] = wmma4(a, b, c[t]);
            }
        }
    }

    const int crow = row0 + ((lane >> 4) << 3);
#pragma unroll
    for (int t = 0; t < HH / 16; ++t) {
        const int col = t * 16 + (lane & 15);
        const float bv = bias[col];
#pragma unroll
        for (int i = 0; i < 8; ++i)
            out[(size_t)(crow + i) * HH + col] = c[t][i] + bv;
    }
}

// ---------------- output projection (+ final elementwise multiply) ----------------
__global__ void k_output(const float* __restrict__ x, const float* __restrict__ Wo,
                         const float* __restrict__ bo, float* __restrict__ outA,
                         float* __restrict__ dout, int branch) {
    int n = blockIdx.x * blockDim.x + threadIdx.x;
    if (n >= NN) return;
    float a0 = bo[0], a1 = bo[1], a2 = bo[2];
    const float* row = x + (size_t)n * HH;
#pragma unroll 8
    for (int k = 0; k < HH; ++k) {
        const float v = row[k];
        a0 += v * Wo[k * OO + 0];
        a1 += v * Wo[k * OO + 1];
        a2 += v * Wo[k * OO + 2];
    }
    if (branch == 0) {
        outA[n * OO + 0] = a0; outA[n * OO + 1] = a1; outA[n * OO + 2] = a2;
    } else {
        dout[n * OO + 0] = outA[n * OO + 0] * a0;
        dout[n * OO + 1] = outA[n * OO + 1] * a1;
        dout[n * OO + 2] = outA[n * OO + 2] * a2;
    }
}

extern "C" void kernel_launch(void* const* d_in, const int* in_sizes, int n_in,
                              void* d_out, int out_size, void* d_ws, size_t ws_size,
                              hipStream_t stream) {
    const float* mf   = (const float*)d_in[0];
    const float* f    = (const float*)d_in[1];
    const float* W1   = (const float*)d_in[2];
    const float* b1   = (const float*)d_in[3];
    const float* rg1w = (const float*)d_in[4];
    const float* rg1r = (const float*)d_in[5];
    const float* rg1b = (const float*)d_in[6];
    const float* rg2w = (const float*)d_in[7];
    const float* rg2r = (const float*)d_in[8];
    const float* rg2b = (const float*)d_in[9];
    const float* Wo   = (const float*)d_in[10];
    const float* bo   = (const float*)d_in[11];
    const int*   ei   = (const int*)d_in[12];
    const int*   et   = (const int*)d_in[13];
    const int* src = ei;            // edge_index[0]
    const int* dst = ei + EE;       // edge_index[1]
    float* out = (float*)d_out;

    float* ws = (float*)d_ws;
    const size_t NH  = (size_t)NN * HH;
    const size_t PHH = (size_t)(HH / 2) * HH;      // packed H x H size in float2
    float*  h    = ws;                              // [N,H]
    float*  hn   = ws + NH;                         // [N,H]
    float*  agg  = ws + 2 * NH;                     // [2,N,H]
    float*  cnt  = ws + 4 * NH;                     // [2,N] -> reciprocals
    float*  outA = ws + 4 * NH + 2 * NN;            // [N,O]
    float2* pk   = (float2*)(ws + 4 * NH + 2 * NN + (size_t)NN * OO);
    float2* P1r  = pk;                              // rg1_root packed
    float2* P1w0 = pk + PHH;                        // rg1_w[0]
    float2* P1w1 = pk + 2 * PHH;                    // rg1_w[1]
    float2* P2r  = pk + 3 * PHH;                    // rg2_root
    float2* P2w0 = pk + 4 * PHH;                    // rg2_w[0]
    float2* P2w1 = pk + 5 * PHH;                    // rg2_w[1]
    float2* PW1  = pk + 6 * PHH;                    // W1 packed [D/2*H]

    // Pack all weight matrices into WMMA B-fragment order (cheap, every call).
    {
        const int npk = (HH / 2) * HH;              // 8192 float2 per HxH
        dim3 pb((npk + 255) / 256), pt(256);
        k_pack<<<pb, pt, 0, stream>>>(rg1r, P1r, HH);
        k_pack<<<pb, pt, 0, stream>>>(rg1w, P1w0, HH);
        k_pack<<<pb, pt, 0, stream>>>(rg1w + HH * HH, P1w1, HH);
        k_pack<<<pb, pt, 0, stream>>>(rg2r, P2r, HH);
        k_pack<<<pb, pt, 0, stream>>>(rg2w, P2w0, HH);
        k_pack<<<pb, pt, 0, stream>>>(rg2w + HH * HH, P2w1, HH);
        const int npk1 = (DD / 2) * HH;             // 1024 float2 for W1
        k_pack<<<(npk1 + 255) / 256, 256, 0, stream>>>(W1, PW1, DD);
    }

    // Per-(relation,dst) edge counts -> reciprocals (shared by both branches)
    hipMemsetAsync(cnt, 0, (size_t)2 * NN * sizeof(float), stream);
    k_count<<<(EE + 255) / 256, 256, 0, stream>>>(dst, et, cnt);
    k_inv<<<(2 * NN + 255) / 256, 256, 0, stream>>>(cnt);

    const int RB  = NN / 16;   // 6250 row blocks of 16 nodes
    const int WPB = 4;         // waves per block
    dim3 gblk((RB + WPB - 1) / WPB), gth(WPB * 32);
    dim3 sblk((EE + 7) / 8), sth(256);   // 8 edge-waves per block

    for (int b = 0; b < 2; ++b) {
        k_input<<<gblk, gth, 0, stream>>>(mf, f, PW1, b1, h, b);

        // conv1: h -> hn
        hipMemsetAsync(agg, 0, 2 * NH * sizeof(float), stream);
        k_scatter<<<sblk, sth, 0, stream>>>(h, src, dst, et, agg);
        k_conv<<<gblk, gth, 0, stream>>>(h, agg, P1r, P1w0, P1w1, rg1b, cnt, hn);

        // conv2: hn -> h
        hipMemsetAsync(agg, 0, 2 * NH * sizeof(float), stream);
        k_scatter<<<sblk, sth, 0, stream>>>(hn, src, dst, et, agg);
        k_conv<<<gblk, gth, 0, stream>>>(hn, agg, P2r, P2w0, P2w1, rg2b, cnt, h);

        k_output<<<(NN + 255) / 256, 256, 0, stream>>>(h, Wo, bo, outA, out, b);
    }
}